// SemanticGuidedOT_76776835383863
// MI455X (gfx1250) — compile-verified
//
#include <hip/hip_runtime.h>
#include <math.h>

#define BB   2
#define CC   256
#define HH   64
#define WW   64
#define NN   4096          // H*W
#define TDD  512
#define KH   128           // C/2
#define EPSI 0.05f
#define LDP  (CC + 4)      // padded LDS row stride (260): 260%64==4 -> conflict-free

typedef __attribute__((ext_vector_type(2))) float v2f;
typedef __attribute__((ext_vector_type(8))) float v8f;

#if defined(__AMDGCN__)
#  if __has_builtin(__builtin_amdgcn_global_load_async_to_lds_b64)
#    define HAVE_ASYNC 1
#  endif
#endif
#ifndef HAVE_ASYNC
#  define HAVE_ASYNC 0
#endif

#if HAVE_ASYNC
typedef int v2i_vs __attribute__((vector_size(8)));     // matches builtin param type
#define GAS __attribute__((address_space(1)))           // prints as __device__
#define LAS __attribute__((address_space(3)))           // prints as __shared__
#endif

// ---------------- FiLM: txt_emb = txt @ Wt^T + bt ----------------
__global__ void k_txt(const float* __restrict__ txt, const float* __restrict__ Wt,
                      const float* __restrict__ bt, float* __restrict__ emb) {
    int t = blockIdx.x * blockDim.x + threadIdx.x;      // 0..B*C-1
    int b = t / CC, c = t % CC;
    const float* x = txt + b * TDD;
    const float* w = Wt + (size_t)c * TDD;
    float acc = bt[c];
#pragma unroll 4
    for (int i = 0; i < TDD; ++i) acc += x[i] * w[i];
    emb[t] = acc;
}

// ---------------- gamma/beta ----------------
__global__ void k_gb(const float* __restrict__ emb, const float* __restrict__ Wg,
                     const float* __restrict__ bg, const float* __restrict__ Wb,
                     const float* __restrict__ bbv, float* __restrict__ gamma,
                     float* __restrict__ beta) {
    int t = blockIdx.x * blockDim.x + threadIdx.x;
    int b = t / CC, c = t % CC;
    const float* e  = emb + b * CC;
    const float* wg = Wg + (size_t)c * CC;
    const float* wb = Wb + (size_t)c * CC;
    float ag = bg[c], ab = bbv[c];
#pragma unroll 4
    for (int i = 0; i < CC; ++i) { ag += e[i] * wg[i]; ab += e[i] * wb[i]; }
    gamma[t] = ag; beta[t] = ab;
}

// ---------------- mod = img*(1+gamma)+beta, transposed to [B,N,C] ----------------
__global__ void k_mod(const float* __restrict__ img, const float* __restrict__ gamma,
                      const float* __restrict__ beta, float* __restrict__ sf) {
    size_t t = (size_t)blockIdx.x * 256 + threadIdx.x;  // b*N*C + n*C + c
    int c = (int)(t % CC);
    size_t r = t / CC;
    int n = (int)(r % NN);
    int b = (int)(r / NN);
    float g = gamma[b * CC + c], be = beta[b * CC + c];
    float x = img[((size_t)b * CC + c) * NN + n];
    sf[t] = x * (1.0f + g) + be;
}

// ---------------- L2-normalize rows (one wave per row) ----------------
__global__ void k_norm(const float* __restrict__ sf, float* __restrict__ sn) {
    int wave = threadIdx.x >> 5, lane = threadIdx.x & 31;
    size_t row = (size_t)blockIdx.x * 8 + wave;
    const float* src = sf + row * CC;
    float s = 0.f;
#pragma unroll
    for (int i = lane; i < CC; i += 32) { float x = src[i]; s += x * x; }
#pragma unroll
    for (int o = 16; o; o >>= 1) s += __shfl_xor(s, o, 32);
    float inv = 1.0f / fmaxf(sqrtf(s), 1e-12f);
    float* dst = sn + row * CC;
#pragma unroll
    for (int i = lane; i < CC; i += 32) dst[i] = src[i] * inv;
}

// ---------------- q normalized; u = 1 ----------------
__global__ void k_q(const float* __restrict__ dens, float* __restrict__ q,
                    float* __restrict__ u) {
    int b = blockIdx.x, tid = threadIdx.x;
    __shared__ float red[256];
    float s = 0.f;
    for (int n = tid; n < NN; n += 256) s += fmaxf(dens[b * NN + n], 0.f) + 1e-6f;
    red[tid] = s; __syncthreads();
    for (int o = 128; o; o >>= 1) { if (tid < o) red[tid] += red[tid + o]; __syncthreads(); }
    float inv = 1.0f / red[0];
    for (int n = tid; n < NN; n += 256) {
        float t = fmaxf(dens[b * NN + n], 0.f) + 1e-6f;
        q[b * NN + n] = t * inv;
        u[b * NN + n] = 1.0f;
    }
}

// ---------------- K = exp((sn sn^T - 1)/eps), fp32 WMMA, 2x2 blocking ----------------
// Block: 8 waves, each wave a 32x32 output tile; A panel (32 rows) staged in LDS
// (async-to-LDS when available), B rows streamed from global/L2.
__global__ void k_simK(const float* __restrict__ sn, float* __restrict__ Kb) {
    __shared__ float As[32 * LDP];                       // 33.3 KB
    int tid  = threadIdx.x;
    int wave = tid >> 5, lane = tid & 31;
    int half = lane >> 4, lrow = lane & 15;
    int j0 = blockIdx.x * 256 + wave * 32;
    int i0 = blockIdx.y * 32;
    int b  = blockIdx.z;
    const float* base = sn + (size_t)b * NN * CC;

    // ---- stage A panel: 32 rows x 256 floats ----
#if HAVE_ASYNC
    for (int idx = tid; idx < 32 * (CC / 2); idx += 256) {
        int row = idx >> 7;                              // 128 float2 per row
        int c2  = (idx & 127) << 1;
        float* src = const_cast<float*>(base + (size_t)(i0 + row) * CC + c2);
        float* dst = &As[row * LDP + c2];
        __builtin_amdgcn_global_load_async_to_lds_b64(
            (GAS v2i_vs*)src, (LAS v2i_vs*)dst, 0, 0);
    }
    asm volatile("s_wait_asynccnt 0x0" ::: "memory");
    __syncthreads();
#else
    for (int idx = tid; idx < 32 * (CC / 2); idx += 256) {
        int row = idx >> 7;
        int c2  = (idx & 127) << 1;
        *(v2f*)&As[row * LDP + c2] =
            *(const v2f*)(base + (size_t)(i0 + row) * CC + c2);
    }
    __syncthreads();
#endif

    const float* a0p = &As[(size_t)lrow * LDP + 2 * half];
    const float* a1p = &As[(size_t)(16 + lrow) * LDP + 2 * half];
    const float* b0p = base + (size_t)(j0 + lrow) * CC + 2 * half;
    const float* b1p = base + (size_t)(j0 + 16 + lrow) * CC + 2 * half;

    v8f acc00 = {}, acc01 = {}, acc10 = {}, acc11 = {};
#pragma unroll 4
    for (int k = 0; k < CC; k += 4) {
        v2f a0 = *(const v2f*)(a0p + k);
        v2f a1 = *(const v2f*)(a1p + k);
        v2f b0 = *(const v2f*)(b0p + k);
        v2f b1 = *(const v2f*)(b1p + k);
        acc00 = __builtin_amdgcn_wmma_f32_16x16x4_f32(false, a0, false, b0, (short)0, acc00, false, false);
        acc01 = __builtin_amdgcn_wmma_f32_16x16x4_f32(false, a0, false, b1, (short)0, acc01, false, false);
        acc10 = __builtin_amdgcn_wmma_f32_16x16x4_f32(false, a1, false, b0, (short)0, acc10, false, false);
        acc11 = __builtin_amdgcn_wmma_f32_16x16x4_f32(false, a1, false, b1, (short)0, acc11, false, false);
    }

    const float ie = 1.0f / EPSI;
    float* outp = Kb + ((size_t)b * NN + i0) * NN + j0;
#pragma unroll
    for (int r = 0; r < 8; ++r) {
        int row0 = r + half * 8, row1 = 16 + row0;
        outp[(size_t)row0 * NN + lrow]      = __expf((acc00[r] - 1.0f) * ie);
        outp[(size_t)row0 * NN + 16 + lrow] = __expf((acc01[r] - 1.0f) * ie);
        outp[(size_t)row1 * NN + lrow]      = __expf((acc10[r] - 1.0f) * ie);
        outp[(size_t)row1 * NN + 16 + lrow] = __expf((acc11[r] - 1.0f) * ie);
    }
}

// ---------------- v = q / (K^T u + 1e-8) ----------------
__global__ void k_vupd(const float* __restrict__ Kb, const float* __restrict__ u,
                       const float* __restrict__ q, float* __restrict__ v) {
    int b   = blockIdx.y;
    int col = blockIdx.x * 64 + (threadIdx.x & 63);
    int grp = threadIdx.x >> 6;
    const float* Kp = Kb + (size_t)b * NN * NN;
    const float* up = u + b * NN;
    float s = 0.f;
    for (int n = grp; n < NN; n += 4) s += Kp[(size_t)n * NN + col] * up[n];
    __shared__ float red[256];
    red[threadIdx.x] = s; __syncthreads();
    if (grp == 0) {
        float t = red[threadIdx.x] + red[threadIdx.x + 64] +
                  red[threadIdx.x + 128] + red[threadIdx.x + 192];
        v[b * NN + col] = q[b * NN + col] / (t + 1e-8f);
    }
}

// ---------------- u = (1/N) / (K v + 1e-8), row sweep with prefetch ----------------
__global__ void k_uupd(const float* __restrict__ Kb, const float* __restrict__ v,
                       float* __restrict__ u) {
    int wave = threadIdx.x >> 5, lane = threadIdx.x & 31;
    int b = blockIdx.y;
    int n = blockIdx.x * 8 + wave;
    const float* Kp = Kb + ((size_t)b * NN + n) * NN;
    const float* vp = v + b * NN;
    float s = 0.f;
    for (int m = lane; m < NN; m += 32) {
        __builtin_prefetch(Kp + m + 512, 0, 0);          // global_prefetch_b8, 2KB ahead
        s += Kp[m] * vp[m];
    }
#pragma unroll
    for (int o = 16; o; o >>= 1) s += __shfl_xor(s, o, 32);
    if (lane == 0) u[b * NN + n] = (1.0f / NN) / (s + 1e-8f);
}

// ---------------- fused[m][c] = v[m] * sum_n (u[n]K[n][m]) sf[n][c], WMMA 2x2 ----------------
// Block: 8 waves; wave w owns c in [32w, 32w+32) (covers all C=256); m-block of 32.
__global__ void k_fused(const float* __restrict__ Kb, const float* __restrict__ sf,
                        const float* __restrict__ u, const float* __restrict__ v,
                        float* __restrict__ fz) {
    __shared__ float ul[NN];                             // 16 KB, u broadcast panel
    int tid  = threadIdx.x;
    int wave = tid >> 5, lane = tid & 31;
    int half = lane >> 4, lrow = lane & 15;
    int c0 = wave * 32;
    int m0 = blockIdx.x * 32;
    int b  = blockIdx.y;
    const float* Kp = Kb + (size_t)b * NN * NN;
    const float* sp = sf + (size_t)b * NN * CC;
    for (int i = tid; i < NN; i += 256) ul[i] = u[b * NN + i];
    __syncthreads();

    v8f acc00 = {}, acc01 = {}, acc10 = {}, acc11 = {};
#pragma unroll 2
    for (int n0 = 0; n0 < NN; n0 += 4) {
        int ka = n0 + 2 * half;
        float u0 = ul[ka], u1 = ul[ka + 1];
        const float* kr0 = Kp + (size_t)ka * NN + m0 + lrow;
        const float* kr1 = Kp + (size_t)(ka + 1) * NN + m0 + lrow;
        v2f a0, a1;
        a0.x = u0 * kr0[0];  a0.y = u1 * kr1[0];
        a1.x = u0 * kr0[16]; a1.y = u1 * kr1[16];
        const float* s0 = sp + (size_t)ka * CC + c0 + lrow;
        const float* s1 = sp + (size_t)(ka + 1) * CC + c0 + lrow;
        v2f b0, b1;
        b0.x = s0[0];  b0.y = s1[0];
        b1.x = s0[16]; b1.y = s1[16];
        acc00 = __builtin_amdgcn_wmma_f32_16x16x4_f32(false, a0, false, b0, (short)0, acc00, false, false);
        acc01 = __builtin_amdgcn_wmma_f32_16x16x4_f32(false, a0, false, b1, (short)0, acc01, false, false);
        acc10 = __builtin_amdgcn_wmma_f32_16x16x4_f32(false, a1, false, b0, (short)0, acc10, false, false);
        acc11 = __builtin_amdgcn_wmma_f32_16x16x4_f32(false, a1, false, b1, (short)0, acc11, false, false);
    }

    const float* vp = v + b * NN;
#pragma unroll
    for (int r = 0; r < 8; ++r) {
        int m0r = m0 + r + half * 8;
        int m1r = m0r + 16;
        float v0 = vp[m0r], v1 = vp[m1r];
        fz[((size_t)b * NN + m0r) * CC + c0 + lrow]      = acc00[r] * v0;
        fz[((size_t)b * NN + m0r) * CC + c0 + 16 + lrow] = acc01[r] * v0;
        fz[((size_t)b * NN + m1r) * CC + c0 + lrow]      = acc10[r] * v1;
        fz[((size_t)b * NN + m1r) * CC + c0 + 16 + lrow] = acc11[r] * v1;
    }
}

// ---------------- out = img + Wo @ fused + bo ----------------
__global__ void k_oproj(const float* __restrict__ img, const float* __restrict__ fz,
                        const float* __restrict__ Wo, const float* __restrict__ bo,
                        float* __restrict__ out) {
    int n = blockIdx.x, b = blockIdx.y, o = threadIdx.x;
    __shared__ float fr[CC];
    fr[o] = fz[((size_t)b * NN + n) * CC + o];
    __syncthreads();
    const float* w = Wo + (size_t)o * CC;
    float s = bo[o];
#pragma unroll 4
    for (int c = 0; c < CC; ++c) s += w[c] * fr[c];
    out[((size_t)b * CC + o) * NN + n] = img[((size_t)b * CC + o) * NN + n] + s;
}

// ---------------- 3x3 conv, C=256 -> 128, SAME ----------------
__global__ void k_conv3(const float* __restrict__ in, const float* __restrict__ Wc1,
                        const float* __restrict__ bc1, float* __restrict__ h1) {
    int tid = threadIdx.x;
    int hw = blockIdx.x * 256 + tid;
    int y = hw >> 6, x = hw & 63;
    int k = blockIdx.y, b = blockIdx.z;
    __shared__ float wl[CC * 9];
    for (int i = tid; i < CC * 9; i += 256) wl[i] = Wc1[(size_t)k * CC * 9 + i];
    __syncthreads();
    float s = bc1[k];
    const float* ip = in + (size_t)b * CC * NN;
    for (int c = 0; c < CC; ++c) {
        const float* icp = ip + (size_t)c * NN;
        const float* wc  = wl + c * 9;
#pragma unroll
        for (int dy = -1; dy <= 1; ++dy) {
            int yy = y + dy; if (yy < 0 || yy >= HH) continue;
#pragma unroll
            for (int dx = -1; dx <= 1; ++dx) {
                int xx = x + dx; if (xx < 0 || xx >= WW) continue;
                s += icp[yy * WW + xx] * wc[(dy + 1) * 3 + (dx + 1)];
            }
        }
    }
    h1[((size_t)b * KH + k) * NN + hw] = s;
}

// ---------------- BN stats ----------------
__global__ void k_bnstat(const float* __restrict__ h1, float* __restrict__ mean,
                         float* __restrict__ invs) {
    int k = blockIdx.x, tid = threadIdx.x;
    __shared__ float rs[256], rs2[256];
    float s = 0.f, s2 = 0.f;
    for (int b = 0; b < BB; ++b) {
        const float* p = h1 + ((size_t)b * KH + k) * NN;
        for (int i = tid; i < NN; i += 256) { float x = p[i]; s += x; s2 += x * x; }
    }
    rs[tid] = s; rs2[tid] = s2; __syncthreads();
    for (int o = 128; o; o >>= 1) {
        if (tid < o) { rs[tid] += rs[tid + o]; rs2[tid] += rs2[tid + o]; }
        __syncthreads();
    }
    if (tid == 0) {
        float m = rs[0] / (float)(BB * NN);
        float var = rs2[0] / (float)(BB * NN) - m * m;
        mean[k] = m; invs[k] = rsqrtf(var + 1e-5f);
    }
}

// ---------------- normalize + ReLU + 1x1 (128 -> 2) ----------------
__global__ void k_heat(const float* __restrict__ h1, const float* __restrict__ mean,
                       const float* __restrict__ invs, const float* __restrict__ bn_g,
                       const float* __restrict__ bn_b, const float* __restrict__ Wc2,
                       const float* __restrict__ bc2, float* __restrict__ heat) {
    int hw = blockIdx.x * 256 + threadIdx.x;
    int b = blockIdx.y;
    float a0 = bc2[0], a1 = bc2[1];
#pragma unroll 4
    for (int k = 0; k < KH; ++k) {
        float x = h1[((size_t)b * KH + k) * NN + hw];
        float r = fmaxf((x - mean[k]) * invs[k] * bn_g[k] + bn_b[k], 0.f);
        a0 += Wc2[k] * r;
        a1 += Wc2[KH + k] * r;
    }
    heat[(size_t)b * 2 * NN + hw]       = a0;
    heat[(size_t)(b * 2 + 1) * NN + hw] = a1;
}

extern "C" void kernel_launch(void* const* d_in, const int* in_sizes, int n_in,
                              void* d_out, int out_size, void* d_ws, size_t ws_size,
                              hipStream_t stream) {
    const float* img  = (const float*)d_in[0];
    const float* txt  = (const float*)d_in[1];
    const float* dens = (const float*)d_in[2];
    const float* Wt   = (const float*)d_in[3];
    const float* bt   = (const float*)d_in[4];
    const float* Wg   = (const float*)d_in[5];
    const float* bg   = (const float*)d_in[6];
    const float* Wb   = (const float*)d_in[7];
    const float* bbv  = (const float*)d_in[8];
    const float* Wo   = (const float*)d_in[9];
    const float* bo   = (const float*)d_in[10];
    const float* Wc1  = (const float*)d_in[11];
    const float* bc1  = (const float*)d_in[12];
    const float* bn_g = (const float*)d_in[13];
    const float* bn_b = (const float*)d_in[14];
    const float* Wc2  = (const float*)d_in[15];
    const float* bc2  = (const float*)d_in[16];

    float* outf = (float*)d_out;                         // [B,C,H,W]
    float* heat = outf + (size_t)BB * CC * NN;           // [B,2,H,W]

    float* w = (float*)d_ws;
    float* emb   = w;  w += BB * CC;
    float* gamma = w;  w += BB * CC;
    float* beta  = w;  w += BB * CC;
    float* sf    = w;  w += (size_t)BB * NN * CC;
    float* sn    = w;  w += (size_t)BB * NN * CC;
    float* q     = w;  w += BB * NN;
    float* u     = w;  w += BB * NN;
    float* v     = w;  w += BB * NN;
    float* fz    = w;  w += (size_t)BB * NN * CC;
    float* h1    = w;  w += (size_t)BB * KH * NN;
    float* mean  = w;  w += KH;
    float* invs  = w;  w += KH;
    float* Kb    = w;  // B*N*N fp32 = 134 MB (L2-resident), largest last

    k_txt<<<dim3((BB * CC) / 256), dim3(256), 0, stream>>>(txt, Wt, bt, emb);
    k_gb <<<dim3((BB * CC) / 256), dim3(256), 0, stream>>>(emb, Wg, bg, Wb, bbv, gamma, beta);
    k_mod<<<dim3(((size_t)BB * NN * CC) / 256), dim3(256), 0, stream>>>(img, gamma, beta, sf);
    k_norm<<<dim3((BB * NN) / 8), dim3(256), 0, stream>>>(sf, sn);
    k_q  <<<dim3(BB), dim3(256), 0, stream>>>(dens, q, u);

    k_simK<<<dim3(NN / 256, NN / 32, BB), dim3(256), 0, stream>>>(sn, Kb);

    for (int it = 0; it < 3; ++it) {
        k_vupd<<<dim3(NN / 64, BB), dim3(256), 0, stream>>>(Kb, u, q, v);
        k_uupd<<<dim3(NN / 8, BB), dim3(256), 0, stream>>>(Kb, v, u);
    }

    k_fused<<<dim3(NN / 32, BB), dim3(256), 0, stream>>>(Kb, sf, u, v, fz);
    k_oproj<<<dim3(NN, BB), dim3(256), 0, stream>>>(img, fz, Wo, bo, outf);

    k_conv3<<<dim3(NN / 256, KH, BB), dim3(256), 0, stream>>>(outf, Wc1, bc1, h1);
    k_bnstat<<<dim3(KH), dim3(256), 0, stream>>>(h1, mean, invs);
    k_heat<<<dim3(NN / 256, BB), dim3(256), 0, stream>>>(h1, mean, invs, bn_g, bn_b,
                                                         Wc2, bc2, heat);
}